// VectorQuantizer_78529182040261
// MI455X (gfx1250) — compile-verified
//
#include <hip/hip_runtime.h>
#include <hip/hip_bf16.h>

// ---------------------------------------------------------------------------
// VectorQuantizer forward for MI455X (gfx1250, wave32, WMMA + async-LDS DMA).
//
// score[n,k] = ||e_k||^2 - 2 * z_n . e_k        (||z_n||^2 dropped: row-const)
// idx[n]     = argmin_k score[n,k]  (first-occurrence ties, like jnp.argmin)
// z_q[n,:]   = weight[idx[n],:]     (straight-through fwd == gather)
// perplexity = exp(-sum(p*log(p+eps))), p = histogram(idx)/N_TOK
//
// Roofline: ~144MB HBM @ 23.3TB/s ~ 6us floor; f32 WMMA (16x16x4) keeps the
// argmin exact while compute stays far below that floor. Codebook tiles are
// DMA'd global->LDS with GLOBAL_LOAD_ASYNC_TO_LDS_B128 (ASYNCcnt), double-
// buffered so the next 64KB tile streams while the current one is computed.
// ---------------------------------------------------------------------------

#define N_TOK   32768
#define E_DIM   512
#define N_E     8192
#define LDA     516          // 512 + 4 floats pad -> conflict-free ds_load_b64
#define M_BLK   64           // tokens per block (4 M-waves x 16)
#define N_STEP  32           // codes per step   (2 N-waves x 16)
#define STEPS   (N_E / N_STEP)

typedef __attribute__((ext_vector_type(2))) float v2f;
typedef __attribute__((ext_vector_type(8))) float v8f;

// ---- gfx1250 async global->LDS DMA (bypasses VGPRs; tracked by ASYNCcnt) --
__device__ __forceinline__ void async_b128(const void* gptr, void* lptr) {
  unsigned lds_addr = (unsigned)(uintptr_t)lptr;           // LDS offset in [31:0]
  unsigned long long gaddr = (unsigned long long)(uintptr_t)gptr;
  asm volatile("global_load_async_to_lds_b128 %0, %1, off"
               :: "v"(lds_addr), "v"(gaddr) : "memory");
}
__device__ __forceinline__ void wait_async0() {
  asm volatile("s_wait_asynccnt 0x0" ::: "memory");
}

// wave32 xor-swizzle (group-of-32 encoding: offset = xor<<10 | or<<5 | and)
#define SWZ_I(x, MASK) __builtin_amdgcn_ds_swizzle((x), (((MASK) << 10) | 0x1f))
#define SWZ_F(x, MASK) __builtin_bit_cast(float, SWZ_I(__builtin_bit_cast(int, (x)), MASK))

#define SWZ_STEP(MASK)                                                        \
  _Pragma("unroll")                                                           \
  for (int i = 0; i < 8; ++i) {                                               \
    float ov = SWZ_F(best[i], MASK);                                          \
    int   oi = SWZ_I(bidx[i], MASK);                                          \
    if (ov < best[i] || (ov == best[i] && oi < bidx[i])) {                    \
      best[i] = ov; bidx[i] = oi;                                             \
    }                                                                         \
  }

// --- kernel 0: zero histogram + precompute ||e_k||^2 -----------------------
__global__ __launch_bounds__(256) void vq_prep(const float4* __restrict__ w4,
                                               float* __restrict__ wnorm,
                                               unsigned int* __restrict__ counts) {
  int i = blockIdx.x * 256 + threadIdx.x;
  if (i < N_E) {
    counts[i] = 0u;
    const float4* row = w4 + (size_t)i * (E_DIM / 4);
    float s = 0.f;
#pragma unroll 8
    for (int j = 0; j < E_DIM / 4; ++j) {
      float4 v = row[j];
      s = fmaf(v.x, v.x, s);
      s = fmaf(v.y, v.y, s);
      s = fmaf(v.z, v.z, s);
      s = fmaf(v.w, v.w, s);
    }
    wnorm[i] = s;
  }
}

// --- kernel 1: fused distance-WMMA + argmin + gather + histogram -----------
__global__ __launch_bounds__(256) void vq_main(const float* __restrict__ z,
                                               const float* __restrict__ w,
                                               const float* __restrict__ wnorm,
                                               unsigned int* __restrict__ counts,
                                               float* __restrict__ zq) {
  extern __shared__ float smem[];
  float* sA      = smem;                        // 64 x LDA  (tokens, resident)
  float* sB0     = smem + M_BLK * LDA;          // 32 x LDA  (codes, buf 0)
  float* sB1     = sB0 + N_STEP * LDA;          // 32 x LDA  (codes, buf 1)
  float* red_val = sB1 + N_STEP * LDA;          // 64 x 2
  int*   red_idx = (int*)(red_val + 2 * M_BLK);
  int*   fidx    = red_idx + 2 * M_BLK;         // 64

  const int tid  = threadIdx.x;
  const int lane = tid & 31;
  const int wave = tid >> 5;     // 0..7
  const int wm   = wave >> 1;    // 0..3 : which 16-token tile
  const int wn   = wave & 1;     // 0..1 : which 16-code tile in the step
  const int half = lane >> 4;    // 0..1
  const int r16  = lane & 15;
  const int tok0 = blockIdx.x * M_BLK;

  const float4* z4 = (const float4*)z;
  const float4* w4 = (const float4*)w;

  // DMA the 64-token A panel (resident all kernel) + first code tile into LDS.
  // Per wave: 32 + 16 outstanding asyncs, under the 63-entry ASYNCcnt limit.
  for (int j = tid; j < M_BLK * (E_DIM / 4); j += 256) {
    int row = j >> 7, c4 = j & 127;
    async_b128(&z4[(size_t)(tok0 + row) * (E_DIM / 4) + c4],
               &sA[row * LDA + c4 * 4]);
  }
  for (int j = tid; j < N_STEP * (E_DIM / 4); j += 256) {
    int row = j >> 7, c4 = j & 127;
    async_b128(&w4[(size_t)row * (E_DIM / 4) + c4], &sB0[row * LDA + c4 * 4]);
  }

  float best[8];
  int   bidx[8];
#pragma unroll
  for (int i = 0; i < 8; ++i) { best[i] = 3.402823466e38f; bidx[i] = 0; }

  // WMMA fragment pointers: both A(16x4) and B(4x16) fragments load
  // (row = lane&15, cols k+2*(lane>>4)) as float2 from row-major LDS.
  const float* pA = &sA[(wm * 16 + r16) * LDA + 2 * half];
  const float* pB0 = &sB0[(wn * 16 + r16) * LDA + 2 * half];
  const float* pB1 = &sB1[(wn * 16 + r16) * LDA + 2 * half];

  for (int step = 0; step < STEPS; ++step) {
    const int nb = step * N_STEP;

    // my DMAs for buffer step&1 (and A at step 0) are done; barrier makes
    // every wave's DMAs visible and retires all readers of buffer (step+1)&1.
    wait_async0();
    __syncthreads();

    // kick off DMA of the next 64KB code tile into the other buffer; it
    // streams from L2 while we run the WMMA chain below.
    if (step + 1 < STEPS) {
      float* nxt = ((step + 1) & 1) ? sB1 : sB0;
      for (int j = tid; j < N_STEP * (E_DIM / 4); j += 256) {
        int row = j >> 7, c4 = j & 127;
        async_b128(&w4[(size_t)(nb + N_STEP + row) * (E_DIM / 4) + c4],
                   &nxt[row * LDA + c4 * 4]);
      }
    }

    const float* pB = (step & 1) ? pB1 : pB0;
    v8f c = {0.f, 0.f, 0.f, 0.f, 0.f, 0.f, 0.f, 0.f};
#pragma unroll 8
    for (int k = 0; k < E_DIM; k += 4) {
      v2f a = *(const v2f*)(pA + k);
      v2f b = *(const v2f*)(pB + k);
      c = __builtin_amdgcn_wmma_f32_16x16x4_f32(false, a, false, b,
                                                (short)0, c, false, false);
    }

    // lane holds D[M = i + 8*half][N = r16]; fold into running argmin
    const int   code = nb + wn * 16 + r16;
    const float wnv  = wnorm[code];
#pragma unroll
    for (int i = 0; i < 8; ++i) {
      float val = fmaf(-2.f, c[i], wnv);
      if (val < best[i]) { best[i] = val; bidx[i] = code; }  // strict <: first idx wins
    }
  }

  // argmin across the 16 lanes of each half (same M rows, different N cols)
  SWZ_STEP(1) SWZ_STEP(2) SWZ_STEP(4) SWZ_STEP(8)

  if (r16 == 0) {
#pragma unroll
    for (int i = 0; i < 8; ++i) {
      int t = wm * 16 + half * 8 + i;        // local token row
      red_val[t * 2 + wn] = best[i];
      red_idx[t * 2 + wn] = bidx[i];
    }
  }
  __syncthreads();

  // combine the two N-wave halves; histogram
  if (tid < M_BLK) {
    float v0 = red_val[tid * 2 + 0], v1 = red_val[tid * 2 + 1];
    int   i0 = red_idx[tid * 2 + 0], i1 = red_idx[tid * 2 + 1];
    int sel = (v1 < v0 || (v1 == v0 && i1 < i0)) ? i1 : i0;
    fidx[tid] = sel;
    atomicAdd(&counts[sel], 1u);
  }
  __syncthreads();

  // gather winning codebook rows -> z_q (128-bit path)
  float4* zq4 = (float4*)zq;
  for (int j = tid; j < M_BLK * (E_DIM / 4); j += 256) {
    int row = j >> 7, q = j & 127;
    int code = fidx[row];
    zq4[(size_t)(tok0 + row) * (E_DIM / 4) + q] = w4[(size_t)code * (E_DIM / 4) + q];
  }
}

// --- kernel 2: perplexity ---------------------------------------------------
__global__ __launch_bounds__(256) void vq_perp(const unsigned int* __restrict__ counts,
                                               float* __restrict__ out) {
  __shared__ float red[256];
  int tid = threadIdx.x;
  float s = 0.f;
  for (int b = tid; b < N_E; b += 256) {
    float p = (float)counts[b] * (1.0f / (float)N_TOK);
    s += p * logf(p + 1e-10f);
  }
  red[tid] = s;
  __syncthreads();
  for (int off = 128; off > 0; off >>= 1) {
    if (tid < off) red[tid] += red[tid + off];
    __syncthreads();
  }
  if (tid == 0) out[0] = expf(-red[0]);
}

// ---------------------------------------------------------------------------
extern "C" void kernel_launch(void* const* d_in, const int* in_sizes, int n_in,
                              void* d_out, int out_size, void* d_ws, size_t ws_size,
                              hipStream_t stream) {
  (void)in_sizes; (void)n_in; (void)out_size; (void)ws_size;
  const float* z = (const float*)d_in[0];
  const float* w = (const float*)d_in[1];
  float*        zq     = (float*)d_out;
  float*        wnorm  = (float*)d_ws;                 // 8192 f32
  unsigned int* counts = (unsigned int*)d_ws + N_E;    // 8192 u32

  const size_t smem_bytes =
      (size_t)(M_BLK * LDA + 2 * N_STEP * LDA) * sizeof(float)  // A + B0 + B1
      + 2 * M_BLK * sizeof(float)                               // red_val
      + 2 * M_BLK * sizeof(int)                                 // red_idx
      + M_BLK * sizeof(int);                                    // fidx

  // allow >48KB dynamic LDS (idempotent; stream-safe)
  hipFuncSetAttribute((const void*)vq_main,
                      hipFuncAttributeMaxDynamicSharedMemorySize,
                      (int)smem_bytes);

  vq_prep<<<N_E / 256, 256, 0, stream>>>((const float4*)w, wnorm, counts);
  vq_main<<<N_TOK / M_BLK, 256, smem_bytes, stream>>>(z, w, wnorm, counts, zq);
  vq_perp<<<1, 256, 0, stream>>>(counts, zq + (size_t)N_TOK * E_DIM);
}